// WaveGainLayer_19121194402127
// MI455X (gfx1250) — compile-verified
//
#include <hip/hip_runtime.h>
#include <hip/hip_bf16.h>

typedef float v2f __attribute__((ext_vector_type(2)));
typedef float v8f __attribute__((ext_vector_type(8)));

#define C_CH   64
#define F_CH   64
#define N_ORI  6

// ---------------------------------------------------------------------------
// Complex band conv = per-orientation complex GEMM via V_WMMA_F32_16X16X4_F32.
// u: (N, C, L, H, W, 2)   g: (L, 2, F, C)   v: (N, F, L, H, W, 2)
// Block: 128 threads (4 wave32); wave w owns F-tile [16w, 16w+16).
// blockIdx = (w_tile, h_block, n*L + l); each block iterates rowsPerBlock h's.
// ---------------------------------------------------------------------------
__global__ __launch_bounds__(128)
void band_conv_wmma(const float* __restrict__ u, const float* __restrict__ g,
                    float* __restrict__ v, int H, int W, int rowsPerBlock)
{
    const int lane = threadIdx.x & 31;
    const int wave = threadIdx.x >> 5;      // 0..3 -> F tile
    const int f0   = wave * 16;
    const int col  = lane & 15;             // N-column / M-row within tile
    const int khal = lane >> 4;             // which K-half this lane holds

    const int wt = blockIdx.x;              // w tile (16 wide)
    const int hb = blockIdx.y;              // h block
    const int nl = blockIdx.z;
    const int n  = nl / N_ORI;
    const int l  = nl % N_ORI;

    // ---- Preload A fragments (weights) into registers, weight-stationary ----
    // A 16x4 layout: lane<16 holds (K=0,K=1) of row M=lane; lane>=16 holds (K=2,K=3).
    const float* gr = g + (size_t)(l * 2 + 0) * F_CH * C_CH;
    const float* gi = g + (size_t)(l * 2 + 1) * F_CH * C_CH;
    const int arow = f0 + col;
    v2f Ar[16], Ai[16];
#pragma unroll
    for (int k = 0; k < 16; ++k) {
        const int c0 = 4 * k + 2 * khal;
        Ar[k] = *(const v2f*)(gr + (size_t)arow * C_CH + c0);
        Ai[k] = *(const v2f*)(gi + (size_t)arow * C_CH + c0);
    }

    // u[n, c, l, h, w, ri] = ubase + c*cstride + (h*W + w)*2 + ri
    const size_t cstride = (size_t)N_ORI * H * W * 2;
    const float* ubase   = u + ((size_t)n * C_CH * N_ORI + l) * (size_t)H * W * 2;
    const int    w0      = wt * 16;

    for (int hr = 0; hr < rowsPerBlock; ++hr) {
        const int h = hb * rowsPerBlock + hr;
        const float* up = ubase + ((size_t)h * W + (w0 + col)) * 2;

        v8f accA = {};   // sum Wr*Ur
        v8f accB = {};   // sum Wi*Ui
        v8f accI = {};   // sum Wr*Ui + Wi*Ur
#pragma unroll
        for (int k = 0; k < 16; ++k) {
            const int c0 = 4 * k + 2 * khal;
            // Each lane loads (re,im) for its column at channels c0 and c0+1:
            // two fully-coalesced 128B segments per global_load_b64.
            v2f ua = *(const v2f*)(up + (size_t)c0       * cstride);
            v2f ub = *(const v2f*)(up + (size_t)(c0 + 1) * cstride);
            v2f br = { ua.x, ub.x };   // B frag, real part (K=c0,c0+1 | c0+2,c0+3)
            v2f bi = { ua.y, ub.y };   // B frag, imag part
            accA = __builtin_amdgcn_wmma_f32_16x16x4_f32(false, Ar[k], false, br,
                                                         (short)0, accA, false, false);
            accB = __builtin_amdgcn_wmma_f32_16x16x4_f32(false, Ai[k], false, bi,
                                                         (short)0, accB, false, false);
            accI = __builtin_amdgcn_wmma_f32_16x16x4_f32(false, Ar[k], false, bi,
                                                         (short)0, accI, false, false);
            accI = __builtin_amdgcn_wmma_f32_16x16x4_f32(false, Ai[k], false, br,
                                                         (short)0, accI, false, false);
        }

        // D layout: lane<16: VGPR r = D[M=r, N=lane]; lane>=16: D[M=8+r, N=lane-16]
#pragma unroll
        for (int r = 0; r < 8; ++r) {
            const int f = f0 + r + 8 * khal;
            v2f o;
            o.x = accA[r] - accB[r];   // real
            o.y = accI[r];             // imag
            float* vp = v + (((((size_t)n * F_CH + f) * N_ORI + l) * H + h) * W
                             + (w0 + col)) * 2;
            *(v2f*)vp = o;
        }
    }
}

// ---------------------------------------------------------------------------
// 3x3 fp32 conv, pad 1. One block per (n, h) output row; 256 threads =
// 32 w-lanes x 8 f-groups (8 f each). Input halo rows staged in LDS.
// u: (N, 64, 32, 32)  w: (64, 64, 3, 3)  out: (N, 64, 32, 32)
// ---------------------------------------------------------------------------
__global__ __launch_bounds__(256)
void lp_conv3x3(const float* __restrict__ u, const float* __restrict__ wgt,
                float* __restrict__ out)
{
    __shared__ float ls[16][3][34];        // 16 channels x 3 rows x (32+2) padded

    const int t     = threadIdx.x;
    const int wx    = t & 31;              // output w
    const int fbase = (t >> 5) * 8;        // 8 output channels per thread
    const int n     = blockIdx.x >> 5;     // H = 32
    const int h     = blockIdx.x & 31;

    float acc[8];
#pragma unroll
    for (int j = 0; j < 8; ++j) acc[j] = 0.0f;

    for (int cc = 0; cc < C_CH; cc += 16) {
        __syncthreads();
        // stage 16 channels x 3 rows (h-1..h+1) x 34 (zero-padded) floats
        for (int idx = t; idx < 16 * 3 * 34; idx += 256) {
            const int cl  = idx / (3 * 34);
            const int rem = idx % (3 * 34);
            const int r   = rem / 34;
            const int x   = rem % 34;
            const int hs  = h + r - 1;
            const int ws  = x - 1;
            float val = 0.0f;
            if ((unsigned)hs < 32u && (unsigned)ws < 32u)
                val = u[(((size_t)n * C_CH + cc + cl) * 32 + hs) * 32 + ws];
            ls[cl][r][x] = val;
        }
        __syncthreads();

        for (int cl = 0; cl < 16; ++cl) {
#pragma unroll
            for (int kh = 0; kh < 3; ++kh) {
                const float u0 = ls[cl][kh][wx + 0];
                const float u1 = ls[cl][kh][wx + 1];
                const float u2 = ls[cl][kh][wx + 2];
#pragma unroll
                for (int j = 0; j < 8; ++j) {
                    // wave-uniform weight address -> broadcast from cache
                    const float* wp = wgt + (((size_t)(fbase + j) * C_CH + cc + cl) * 3
                                             + kh) * 3;
                    acc[j] = fmaf(wp[0], u0,
                             fmaf(wp[1], u1,
                             fmaf(wp[2], u2, acc[j])));
                }
            }
        }
    }

#pragma unroll
    for (int j = 0; j < 8; ++j)
        out[(((size_t)n * C_CH + fbase + j) * 32 + h) * 32 + wx] = acc[j];
}

// ---------------------------------------------------------------------------
extern "C" void kernel_launch(void* const* d_in, const int* in_sizes, int n_in,
                              void* d_out, int out_size, void* d_ws, size_t ws_size,
                              hipStream_t stream)
{
    const float* u_lp = (const float*)d_in[0];   // (16, 64, 32, 32)
    const float* u0   = (const float*)d_in[1];   // (16, 64, 6, 64, 64, 2)
    const float* u1   = (const float*)d_in[2];   // (16, 64, 6, 32, 32, 2)
    const float* g_lp = (const float*)d_in[3];   // (64, 64, 3, 3)
    const float* g0   = (const float*)d_in[4];   // (6, 2, 64, 64)
    const float* g1   = (const float*)d_in[5];   // (6, 2, 64, 64)

    float* v_lp = (float*)d_out;                                   // 1,048,576
    float* v0   = v_lp + (size_t)16 * 64 * 32 * 32;                // 50,331,648
    float* v1   = v0   + (size_t)16 * 64 * 6 * 64 * 64 * 2;        // 12,582,912

    (void)in_sizes; (void)n_in; (void)out_size; (void)d_ws; (void)ws_size;

    // v_lp: 3x3 conv, one block per (n, h)
    lp_conv3x3<<<dim3(16 * 32), 256, 0, stream>>>(u_lp, g_lp, v_lp);

    // v0: H=W=64 -> w tiles=4, h blocks=8 (8 rows each), z = N*L = 96
    band_conv_wmma<<<dim3(4, 8, 96), 128, 0, stream>>>(u0, g0, v0, 64, 64, 8);

    // v1: H=W=32 -> w tiles=2, h blocks=4 (8 rows each), z = 96
    band_conv_wmma<<<dim3(2, 4, 96), 128, 0, stream>>>(u1, g1, v1, 32, 32, 8);
}